// NCEAverage_19292993094131
// MI455X (gfx1250) — compile-verified
//
#include <hip/hip_runtime.h>
#include <hip/hip_bf16.h>

typedef __attribute__((ext_vector_type(2))) float v2f;
typedef __attribute__((ext_vector_type(8))) float v8f;

#define BATCH   256
#define DIM     128
#define NROWS   1000000
#define KP1     2048
#define INV_T   (1.0f / 0.07f)

// ---------------------------------------------------------------------------
// Scoring: out_l[b,k]  = dot(memory_ab[idx[b,k]], l[b])  / T
//          out_ab[b,k] = dot(memory_l [idx[b,k]], ab[b]) / T
// One block per sample b (8 waves). Each wave handles 16 tiles of 16 k-rows.
// Full-f32 matrix math via V_WMMA_F32_16X16X4_F32 (D-tile columns all equal
// since B is the broadcast vector).
// ---------------------------------------------------------------------------
__global__ __launch_bounds__(256) void nce_score_kernel(
    const float* __restrict__ l, const float* __restrict__ ab,
    const int*   __restrict__ idx,
    const float* __restrict__ memory_l, const float* __restrict__ memory_ab,
    float* __restrict__ out_l, float* __restrict__ out_ab)
{
    __shared__ __align__(16) float sl[DIM];
    __shared__ __align__(16) float sab[DIM];

    const int b   = blockIdx.x;
    const int tid = threadIdx.x;

    // cooperative load of the two per-sample vectors into LDS
    if (tid < DIM) sl[tid]        = l [b * DIM + tid];
    else           sab[tid - DIM] = ab[b * DIM + (tid - DIM)];
    __syncthreads();

    const int lane = tid & 31;
    const int wave = tid >> 5;
    const int m    = lane & 15;   // A-matrix row within tile
    const int hi   = lane >> 4;   // 0: K={0,1} of quad, 1: K={2,3}

    for (int t = 0; t < 16; ++t) {
        const int tile = wave * 16 + t;      // 0..127
        const int k0   = tile * 16;
        const int r    = idx[b * KP1 + k0 + m];
        const float* rowL  = memory_l  + (size_t)r * DIM;
        const float* rowAB = memory_ab + (size_t)r * DIM;

        v8f c_ab = {};   // scores of memory_l rows vs ab
        v8f c_l  = {};   // scores of memory_ab rows vs l

#pragma unroll
        for (int kk = 0; kk < 32; ++kk) {
            const int d = kk * 4 + hi * 2;
            // gathered rows: single-use -> non-temporal
            v2f aL  = __builtin_nontemporal_load((const v2f*)(rowL  + d));
            v2f aAB = __builtin_nontemporal_load((const v2f*)(rowAB + d));
            // broadcast-vector B fragments (same per-lane K indexing as A)
            v2f bAB = *(const v2f*)(&sab[d]);
            v2f bL  = *(const v2f*)(&sl[d]);
            c_ab = __builtin_amdgcn_wmma_f32_16x16x4_f32(
                false, aL,  false, bAB, (short)0, c_ab, false, false);
            c_l  = __builtin_amdgcn_wmma_f32_16x16x4_f32(
                false, aAB, false, bL,  (short)0, c_l,  false, false);
        }

        // Every column of the D tile is identical. Lane 0 (N=0) holds rows
        // 0..7 in its 8 accumulator VGPRs, lane 16 holds rows 8..15.
        if (m == 0) {
#pragma unroll
            for (int j = 0; j < 8; ++j) {
                const int row = hi * 8 + j;
                out_l [b * KP1 + k0 + row] = c_l[j]  * INV_T;
                out_ab[b * KP1 + k0 + row] = c_ab[j] * INV_T;
            }
        }
    }
}

// ---------------------------------------------------------------------------
// EMA + L2-normalize the 256 touched rows of each bank (runs after the bulk
// memcpy so it overwrites the stale copies). One wave per (sample, bank).
// ---------------------------------------------------------------------------
__global__ __launch_bounds__(64) void nce_update_kernel(
    const float* __restrict__ l, const float* __restrict__ ab,
    const int*   __restrict__ y,
    const float* __restrict__ memory_l, const float* __restrict__ memory_ab,
    float* __restrict__ new_memory_l, float* __restrict__ new_memory_ab)
{
    const int b    = blockIdx.x;
    const int wave = threadIdx.x >> 5;   // 0 -> l bank, 1 -> ab bank
    const int lane = threadIdx.x & 31;
    const int row  = y[b];

    const float* mem  = (wave == 0) ? memory_l     : memory_ab;
    const float* x    = (wave == 0) ? l            : ab;
    float*       outm = (wave == 0) ? new_memory_l : new_memory_ab;

    const float4 mv = *(const float4*)(mem + (size_t)row * DIM + lane * 4);
    const float4 xv = *(const float4*)(x   + (size_t)b   * DIM + lane * 4);

    float4 p;
    p.x = 0.5f * mv.x + 0.5f * xv.x;
    p.y = 0.5f * mv.y + 0.5f * xv.y;
    p.z = 0.5f * mv.z + 0.5f * xv.z;
    p.w = 0.5f * mv.w + 0.5f * xv.w;

    float ss = p.x * p.x + p.y * p.y + p.z * p.z + p.w * p.w;
#pragma unroll
    for (int off = 16; off > 0; off >>= 1)
        ss += __shfl_xor(ss, off, 32);

    const float inv = 1.0f / sqrtf(ss);
    float4 o;
    o.x = p.x * inv; o.y = p.y * inv; o.z = p.z * inv; o.w = p.w * inv;
    *(float4*)(outm + (size_t)row * DIM + lane * 4) = o;
}

// ---------------------------------------------------------------------------
extern "C" void kernel_launch(void* const* d_in, const int* in_sizes, int n_in,
                              void* d_out, int out_size, void* d_ws, size_t ws_size,
                              hipStream_t stream) {
    const float* l      = (const float*)d_in[0];
    const float* ab     = (const float*)d_in[1];
    const int*   y      = (const int*)  d_in[2];
    const int*   idx    = (const int*)  d_in[3];
    const float* mem_l  = (const float*)d_in[4];
    const float* mem_ab = (const float*)d_in[5];

    float* out    = (float*)d_out;
    float* out_l  = out;                                   // [B, K1, 1]
    float* out_ab = out + (size_t)BATCH * KP1;             // [B, K1, 1]
    float* new_l  = out + 2 * (size_t)BATCH * KP1;         // [N, D]
    float* new_ab = new_l + (size_t)NROWS * DIM;           // [N, D]

    // 1) gather + WMMA scoring (independent of bank updates)
    nce_score_kernel<<<BATCH, 256, 0, stream>>>(
        l, ab, idx, mem_l, mem_ab, out_l, out_ab);

    // 2) bulk copy of both banks at peak bandwidth
    hipMemcpyAsync(new_l,  mem_l,  (size_t)NROWS * DIM * sizeof(float),
                   hipMemcpyDeviceToDevice, stream);
    hipMemcpyAsync(new_ab, mem_ab, (size_t)NROWS * DIM * sizeof(float),
                   hipMemcpyDeviceToDevice, stream);

    // 3) overwrite the 256 EMA'd rows (stream-ordered after the copies)
    nce_update_kernel<<<BATCH, 64, 0, stream>>>(
        l, ab, y, mem_l, mem_ab, new_l, new_ab);
}